// GraphEmbeddings_44942537786134
// MI455X (gfx1250) — compile-verified
//
#include <hip/hip_runtime.h>

// ---------------------------------------------------------------------------
// GATv2Conv (H=2, C=32, edge_dim=2, x-dim=1) + Linear(64->10), collapsed.
//
// Exact algebra used:
//   xl[n][j] = x[n]*Wl[j] + bl[j]          (rank-1, x has one feature)
//   sum_e alpha_e = 1 per (dst,head)       (softmax; self-loops => no empty seg)
//   => out[n][j] = Wl[j]*z[n][h(j)] + bl[j] + bias[j],
//      z[n][h]   = sum_{e:dst=n} alpha_e * x[src_e]
//   => d[n][o]   = z0*G0[o] + z1*G1[o] + K[o]     (a [N,4]x[4,16] GEMM)
// The final GEMM maps exactly onto V_WMMA_F32_16X16X4_F32 (fp32, exact).
// ---------------------------------------------------------------------------

typedef __attribute__((ext_vector_type(2))) float v2f;
typedef __attribute__((ext_vector_type(8))) float v8f;

#define HC   64
#define CCH  32
#define OUTD 10
#define NEG_SLOPE 0.2f

// Monotone float<->uint mapping so atomicMax(uint) implements float max.
__device__ __forceinline__ unsigned flipf(float f) {
  unsigned u = __float_as_uint(f);
  return (u & 0x80000000u) ? ~u : (u | 0x80000000u);
}
__device__ __forceinline__ float unflipf(unsigned u) {
  unsigned v = (u & 0x80000000u) ? ~u : (u & 0x7fffffffu);
  // note: inverse of flipf: if MSB set, original was non-negative (u|0x8000..)
  return __uint_as_float((u & 0x80000000u) ? (u & 0x7fffffffu) : ~u);
  (void)v;
}

// --- K0: zero accumulators -------------------------------------------------
__global__ void k_init(unsigned* __restrict__ deg, float* __restrict__ sattr,
                       unsigned* __restrict__ smaxk, float* __restrict__ denom,
                       float* __restrict__ zacc, int N) {
  int n = blockIdx.x * blockDim.x + threadIdx.x;
  if (n < N) {
    deg[n] = 0u;
    sattr[2 * n] = 0.f;  sattr[2 * n + 1] = 0.f;
    smaxk[2 * n] = 0u;   smaxk[2 * n + 1] = 0u;   // 0 == most-negative key
    denom[2 * n] = 0.f;  denom[2 * n + 1] = 0.f;
    zacc[2 * n]  = 0.f;  zacc[2 * n + 1]  = 0.f;
  }
}

// --- K1: in-degree + sum of incoming edge_attr (for self-loop 'mean') ------
__global__ void k_deg(const int* __restrict__ ei, const float* __restrict__ ea,
                      unsigned* __restrict__ deg, float* __restrict__ sattr, int E) {
  int e = blockIdx.x * blockDim.x + threadIdx.x;
  if (e < E) {
    int d = ei[E + e];  // row 1 of [2,E] = dst
    atomicAdd(&deg[d], 1u);
    atomicAdd(&sattr[2 * d],     ea[2 * e]);
    atomicAdd(&sattr[2 * d + 1], ea[2 * e + 1]);
  }
}

// --- K2: attention score per augmented edge + segment max ------------------
__global__ void k_score(const int* __restrict__ ei, const float* __restrict__ ea,
                        const float* __restrict__ x,
                        const float* __restrict__ Wl, const float* __restrict__ bl,
                        const float* __restrict__ Wr, const float* __restrict__ br,
                        const float* __restrict__ We, const float* __restrict__ att,
                        const unsigned* __restrict__ deg, const float* __restrict__ sattr,
                        float* __restrict__ scores, unsigned* __restrict__ smaxk,
                        int E, int N) {
  __shared__ float swl[HC], swr[HC], sb[HC], swe0[HC], swe1[HC], satt[HC];
  int t = threadIdx.x;
  if (t < HC) {
    swl[t]  = Wl[t];
    swr[t]  = Wr[t];
    sb[t]   = bl[t] + br[t];
    swe0[t] = We[t];         // We row 0
    swe1[t] = We[HC + t];    // We row 1
    satt[t] = att[t];        // [H,C] flat == j
  }
  __syncthreads();

  int e = blockIdx.x * blockDim.x + t;
  if (e >= E + N) return;

  int s, d; float ea0, ea1;
  if (e < E) {
    s = ei[e]; d = ei[E + e];
    ea0 = ea[2 * e]; ea1 = ea[2 * e + 1];
  } else {                       // self loop, attr = mean of incoming attrs
    int n = e - E; s = n; d = n;
    float dv = (float)deg[n];
    float m = dv > 1.f ? dv : 1.f;
    ea0 = sattr[2 * n] / m; ea1 = sattr[2 * n + 1] / m;
  }
  float xs = x[s], xd = x[d];
  float a0 = 0.f, a1 = 0.f;
#pragma unroll 8
  for (int c = 0; c < CCH; ++c) {
    int j = c;
    float t0 = fmaf(xs, swl[j], fmaf(xd, swr[j],
               fmaf(ea0, swe0[j], fmaf(ea1, swe1[j], sb[j]))));
    t0 = t0 > 0.f ? t0 : NEG_SLOPE * t0;
    a0 = fmaf(satt[j], t0, a0);
    j = CCH + c;
    float t1 = fmaf(xs, swl[j], fmaf(xd, swr[j],
               fmaf(ea0, swe0[j], fmaf(ea1, swe1[j], sb[j]))));
    t1 = t1 > 0.f ? t1 : NEG_SLOPE * t1;
    a1 = fmaf(satt[j], t1, a1);
  }
  scores[2 * e] = a0;
  scores[2 * e + 1] = a1;
  atomicMax(&smaxk[2 * d],     flipf(a0));
  atomicMax(&smaxk[2 * d + 1], flipf(a1));
}

// --- K3: exp(score - max), fused denom and z accumulation ------------------
__global__ void k_accum(const int* __restrict__ ei, const float* __restrict__ x,
                        const float* __restrict__ scores,
                        const unsigned* __restrict__ smaxk,
                        float* __restrict__ denom, float* __restrict__ zacc,
                        int E, int N) {
  int e = blockIdx.x * blockDim.x + threadIdx.x;
  if (e >= E + N) return;
  int s, d;
  if (e < E) { s = ei[e]; d = ei[E + e]; } else { s = d = e - E; }
  float xs = x[s];
  float m0 = unflipf(smaxk[2 * d]);
  float m1 = unflipf(smaxk[2 * d + 1]);
  float v0 = __expf(scores[2 * e]     - m0);
  float v1 = __expf(scores[2 * e + 1] - m1);
  atomicAdd(&denom[2 * d],     v0);
  atomicAdd(&denom[2 * d + 1], v1);
  atomicAdd(&zacc[2 * d],      v0 * xs);
  atomicAdd(&zacc[2 * d + 1],  v1 * xs);
}

// --- K4: fold Wl/bl/bias/W2/b2 into the tiny 4x16 B-matrix -----------------
// GK rows: [G0 ; G1 ; K ; 0], 16 cols (10 valid + 6 zero pad).
__global__ void k_gk(const float* __restrict__ Wl, const float* __restrict__ bl,
                     const float* __restrict__ bias, const float* __restrict__ W2,
                     const float* __restrict__ b2, float* __restrict__ GK) {
  int o = threadIdx.x;
  if (o < 16) {
    float g0 = 0.f, g1 = 0.f, k = 0.f;
    if (o < OUTD) {
      for (int c = 0; c < CCH; ++c) {
        g0 += Wl[c]       * W2[c * OUTD + o];
        g1 += Wl[CCH + c] * W2[(CCH + c) * OUTD + o];
      }
      for (int j = 0; j < HC; ++j) k += (bl[j] + bias[j]) * W2[j * OUTD + o];
      k += b2[o];
    }
    GK[o] = g0; GK[16 + o] = g1; GK[32 + o] = k; GK[48 + o] = 0.f;
  }
}

// --- K5: final projection via V_WMMA_F32_16X16X4_F32 -----------------------
// A (16x4) per tile: row m = [z0, z1, 1, 0] for node m.
//   lanes 0-15 : VGPR0=A[m][0], VGPR1=A[m][1]   (m = lane)
//   lanes 16-31: VGPR0=A[m][2], VGPR1=A[m][3]   (m = lane-16)
// B (4x16): lanes 0-15: VGPR0=B[0][n], VGPR1=B[1][n]; lanes 16-31: B[2]/B[3].
// D (16x16): lane L holds col L%16; VGPR r holds row r (+8 if L>=16).
__global__ void k_final(const float* __restrict__ zacc, const float* __restrict__ denom,
                        const float* __restrict__ GK, float* __restrict__ out, int N) {
  int lane  = threadIdx.x & 31;
  int wave  = (blockIdx.x * blockDim.x + threadIdx.x) >> 5;
  int nwav  = (gridDim.x * blockDim.x) >> 5;
  int half  = lane >> 4;
  int lcol  = lane & 15;

  v2f b;
  if (!half) { b.x = GK[lcol];      b.y = GK[16 + lcol]; }
  else       { b.x = GK[32 + lcol]; b.y = 0.f; }

  int tiles = (N + 15) >> 4;
  for (int tile = wave; tile < tiles; tile += nwav) {
    int node = (tile << 4) + lcol;
    v2f a;
    if (!half) {
      float z0 = 0.f, z1 = 0.f;
      if (node < N) {
        z0 = zacc[2 * node]     / denom[2 * node];
        z1 = zacc[2 * node + 1] / denom[2 * node + 1];
      }
      a.x = z0; a.y = z1;
    } else {
      a.x = 1.f; a.y = 0.f;        // constant row K=2 (adds the K[o] term)
    }
    v8f c = {};
    v8f d = __builtin_amdgcn_wmma_f32_16x16x4_f32(
        /*neg_a=*/false, a, /*neg_b=*/false, b,
        /*c_mod=*/(short)0, c, /*reuse_a=*/false, /*reuse_b=*/false);

    if (lcol < OUTD) {
      int mbase = (tile << 4) + (half ? 8 : 0);
#pragma unroll
      for (int r = 0; r < 8; ++r) {
        int m = mbase + r;
        if (m < N) out[m * OUTD + lcol] = d[r];
      }
    }
  }
}

// ---------------------------------------------------------------------------
extern "C" void kernel_launch(void* const* d_in, const int* in_sizes, int n_in,
                              void* d_out, int out_size, void* d_ws, size_t ws_size,
                              hipStream_t stream) {
  const float* x    = (const float*)d_in[0];
  const int*   ei   = (const int*)d_in[1];
  const float* ea   = (const float*)d_in[2];
  const float* Wl   = (const float*)d_in[3];
  const float* bl   = (const float*)d_in[4];
  const float* Wr   = (const float*)d_in[5];
  const float* br   = (const float*)d_in[6];
  const float* We   = (const float*)d_in[7];
  const float* att  = (const float*)d_in[8];
  const float* bias = (const float*)d_in[9];
  const float* W2   = (const float*)d_in[10];
  const float* b2   = (const float*)d_in[11];
  float* out = (float*)d_out;

  int N = in_sizes[0];          // 100000
  int E = in_sizes[1] / 2;      // 1600000

  // workspace partition (~18 MB)
  char* w = (char*)d_ws;
  unsigned* deg   = (unsigned*)w;  w += (size_t)N * 4;
  float*    sattr = (float*)w;     w += (size_t)N * 8;
  unsigned* smaxk = (unsigned*)w;  w += (size_t)N * 8;
  float*    denom = (float*)w;     w += (size_t)N * 8;
  float*    zacc  = (float*)w;     w += (size_t)N * 8;
  float*    scores= (float*)w;     w += (size_t)(E + N) * 8;
  float*    GK    = (float*)w;     // 64 floats

  const int T = 256;
  int EN = E + N;
  k_init <<<(N  + T - 1) / T, T, 0, stream>>>(deg, sattr, smaxk, denom, zacc, N);
  k_deg  <<<(E  + T - 1) / T, T, 0, stream>>>(ei, ea, deg, sattr, E);
  k_score<<<(EN + T - 1) / T, T, 0, stream>>>(ei, ea, x, Wl, bl, Wr, br, We, att,
                                              deg, sattr, scores, smaxk, E, N);
  k_accum<<<(EN + T - 1) / T, T, 0, stream>>>(ei, x, scores, smaxk, denom, zacc, E, N);
  k_gk   <<<1, 64, 0, stream>>>(Wl, bl, bias, W2, b2, GK);
  k_final<<<160, T, 0, stream>>>(zacc, denom, GK, out, N);
}